// SumOfGammaNoiseDistribution_33466385171131
// MI455X (gfx1250) — compile-verified
//
#include <hip/hip_runtime.h>

// ---------------------------------------------------------------------------
// Sum-of-Gamma noise: out[e] = (sum_{i=1..10} Gamma(1/k, rate=i/k) - ln 10)/k
//                            = sum_{i=1..10} g_i/i  -  ln(10)/k,
//   where g_i ~ Gamma(1/k) standard.  (k-scaling cancels algebraically.)
//
// Compute-bound elementwise generator (memory floor ~5.5us @ 23.3 TB/s,
// compute is >30x that): Philox4x32-7 counter RNG + Box-Muller +
// Marsaglia-Tsang with boost for alpha<1. All transcendentals are single
// hardware TRANS ops (v_log/v_exp/v_sqrt/v_rsq/v_rcp/v_cos) which co-execute
// with the Philox integer VALU stream; uniform parameters land in CDNA5's
// scalar-float SALU. Output is a write-once 128MB stream -> NT b128 stores.
// ---------------------------------------------------------------------------

typedef __attribute__((ext_vector_type(4))) float v4f;

#define PHILOX_M0 0xD2511F53u
#define PHILOX_M1 0xCD9E8D57u
#define PHILOX_W0 0x9E3779B9u
#define PHILOX_W1 0xBB67AE85u

struct U4 { unsigned x, y, z, w; };

// Philox-4x32-7: stateless keyed counter RNG (7 rounds = Crush-resistant
// minimum). Deterministic per (element, iteration, attempt) -> replay safe.
__device__ __forceinline__ U4 philox7(unsigned c0, unsigned c1,
                                      unsigned c2, unsigned c3) {
  unsigned k0 = 0xA341316Cu, k1 = 0xC8013EA4u;
#pragma unroll
  for (int r = 0; r < 7; ++r) {
    unsigned hi0 = __umulhi(PHILOX_M0, c0), lo0 = PHILOX_M0 * c0;
    unsigned hi1 = __umulhi(PHILOX_M1, c2), lo1 = PHILOX_M1 * c2;
    unsigned n0 = hi1 ^ c1 ^ k0;
    unsigned n2 = hi0 ^ c3 ^ k1;
    c0 = n0; c1 = lo1; c2 = n2; c3 = lo0;
    k0 += PHILOX_W0; k1 += PHILOX_W1;
  }
  return {c0, c1, c2, c3};
}

// u32 -> uniform in (0,1), never 0 (safe for log)
__device__ __forceinline__ float u01(unsigned x) {
  return (float)(x >> 8) * 0x1.0p-24f + 0x1.0p-25f;
}

// natural log via hardware v_log_f32 (log2) * ln2
__device__ __forceinline__ float lnf(float x) {
  return __builtin_amdgcn_logf(x) * 0.6931471805599453f;
}

// One output element's accumulator: sum_{i=1..10} Gamma(1/k) * (1/i).
// d, cc are wave-uniform Marsaglia-Tsang params (shape alpha+1); kv = k is
// the boost exponent (U^(1/alpha) = U^k).
__device__ __forceinline__ float gamma_sum_elem(unsigned idx, float kv,
                                                float d, float cc) {
  float acc = 0.0f;

#pragma unroll 1
  for (int i = 1; i <= 10; ++i) {
    U4 r = philox7(idx, (unsigned)i, 0u, 0x33466385u);
    float bu = u01(r.w);        // boost uniform (from attempt 0 only)
    float dv = d * 0.02f;       // fallback if all 3 attempts reject (~6e-5)

#pragma unroll 1
    for (int a = 0; a < 3; ++a) {
      if (a > 0) r = philox7(idx, (unsigned)i, (unsigned)a, 0x33466385u);
      float uu1 = u01(r.x), uu2 = u01(r.y), uu3 = u01(r.z);
      // Box-Muller normal. v_cos_f32 input is in revolutions (angle/2pi),
      // so cos(2*pi*uu2) is exactly v_cos_f32(uu2) -- the 2*pi mul is free.
      float n = __builtin_amdgcn_sqrtf(-2.0f * lnf(uu1)) *
                __builtin_amdgcn_cosf(uu2);
      float t = fmaf(cc, n, 1.0f);
      float v = t * t * t;
      if (v > 0.0f) {
        // log acceptance: ln(u3) < 0.5*n^2 + d*(1 - v + ln v)
        if (lnf(uu3) < fmaf(0.5f * n, n, d * (1.0f - v + lnf(v)))) {
          dv = d * v;           // Gamma(alpha+1) draw
          break;
        }
      }
    }

    // boost: Gamma(alpha) = Gamma(alpha+1) * U^(1/alpha),  1/alpha == k
    float g = dv * __builtin_amdgcn_exp2f(__builtin_amdgcn_logf(bu) * kv);
    acc += g * __builtin_amdgcn_rcpf((float)i);   // * 1/i  (k cancelled)
  }
  return acc;
}

__global__ void __launch_bounds__(256)
SumOfGammaNoise_kernel(const float* __restrict__ kptr,
                       float* __restrict__ out, int n) {
  const float kv    = kptr[0];
  const float alpha = __builtin_amdgcn_rcpf(kv);        // 1/k
  // Marsaglia-Tsang params for shape alpha+1 (wave-uniform -> scalar SALU)
  const float d  = alpha + 0.66666667f;                 // (alpha+1) - 1/3
  const float cc = __builtin_amdgcn_rsqf(9.0f * d);     // 1/(3*sqrt(d))
  const float bias = 2.302585093f * alpha;              // ln(10)/k

  const int tid  = blockIdx.x * blockDim.x + threadIdx.x;
  const int base = tid * 4;
  if (base >= n) return;

  // Compute all 4 lanes-worth through ONE code instance (no duplicated tail
  // pipeline); clamp indices so tail threads just recompute element n-1.
  v4f o;
#pragma unroll
  for (int j = 0; j < 4; ++j) {
    int e = base + j;
    e = e < n ? e : (n - 1);
    o[j] = gamma_sum_elem((unsigned)e, kv, d, cc) - bias;
  }

  if (base + 3 < n) {
    // write-once 128MB stream: 16B non-temporal store, don't rinse L2
    __builtin_nontemporal_store(o, (v4f*)(out + base));
  } else {
    for (int j = 0; j < 4 && base + j < n; ++j) out[base + j] = o[j];
  }
}

extern "C" void kernel_launch(void* const* d_in, const int* in_sizes, int n_in,
                              void* d_out, int out_size, void* d_ws,
                              size_t ws_size, hipStream_t stream) {
  (void)in_sizes; (void)n_in; (void)d_ws; (void)ws_size;
  const float* kptr = (const float*)d_in[0];   // scalar k (device)
  float* out = (float*)d_out;

  const int n       = out_size;                // rows*cols elements
  const int block   = 256;                     // 8 wave32s per block
  const int threads = (n + 3) / 4;             // 4 elements / thread
  int grid = (threads + block - 1) / block;
  if (grid < 1) grid = 1;

  SumOfGammaNoise_kernel<<<grid, block, 0, stream>>>(kptr, out, n);
}